// MambaBottleneck_29987461661347
// MI455X (gfx1250) — compile-verified
//
#include <hip/hip_runtime.h>
#include <cstddef>
#include <cstdint>

// ---------------- problem constants ----------------
#define BB    4
#define CC    192
#define DD    4
#define HHH   16
#define WWW   16
#define LLQ   1024            // DD*HHH*WWW
#define DIN   384
#define NST   64
#define DTR   12
#define NDBL  140             // DTR + 2*NST
#define MROWS 4096            // BB*LLQ
#define KMAX  384             // max GEMM K

typedef float v2f __attribute__((ext_vector_type(2)));
typedef float v8f __attribute__((ext_vector_type(8)));
typedef unsigned int u32x4 __attribute__((ext_vector_type(4)));
typedef int i32x4 __attribute__((ext_vector_type(4)));
typedef int i32x8 __attribute__((ext_vector_type(8)));

__device__ __forceinline__ float gelu_exact(float x) {
    return 0.5f * x * (1.0f + erff(x * 0.70710678118654752f));
}
__device__ __forceinline__ float sigmoidf_(float x) {
    return 1.0f / (1.0f + expf(-x));
}

// ---------------------------------------------------------------------------
// K1: spatial mean pool + softmax branch gate.  grid=BB, block=CC
// ---------------------------------------------------------------------------
__global__ void pool_gate_kernel(const float* __restrict__ x,
                                 const float* __restrict__ wg_w,
                                 const float* __restrict__ wg_b,
                                 float* __restrict__ pooled,
                                 float* __restrict__ wts) {
    __shared__ float ps[CC];
    __shared__ float lg[3];
    int b = blockIdx.x, c = threadIdx.x;
    const float* p = x + ((size_t)b * CC + c) * LLQ;
    float s = 0.f;
    for (int l = 0; l < LLQ; ++l) s += p[l];
    s *= (1.0f / (float)LLQ);
    ps[c] = s;
    pooled[b * CC + c] = s;
    __syncthreads();
    if (c < 3) {
        float t = wg_b[c];
        for (int ic = 0; ic < CC; ++ic) t += wg_w[c * CC + ic] * ps[ic];
        lg[c] = t;
    }
    __syncthreads();
    if (c == 0) {
        float m = fmaxf(lg[0], fmaxf(lg[1], lg[2]));
        float e0 = expf(lg[0] - m), e1 = expf(lg[1] - m), e2 = expf(lg[2] - m);
        float inv = 1.0f / (e0 + e1 + e2);
        wts[b * 3 + 0] = e0 * inv;
        wts[b * 3 + 1] = e1 * inv;
        wts[b * 3 + 2] = e2 * inv;
    }
}

// ---------------------------------------------------------------------------
// K2: tri-directional 1x1x1 projections, gated; writes x_seq [B,L,C].
// grid = (LLQ, BB), block = CC
// ---------------------------------------------------------------------------
__global__ void triproj_kernel(const float* __restrict__ x,
                               const float* __restrict__ px_w, const float* __restrict__ px_b,
                               const float* __restrict__ py_w, const float* __restrict__ py_b,
                               const float* __restrict__ pz_w, const float* __restrict__ pz_b,
                               const float* __restrict__ wts,
                               float* __restrict__ xseq) {
    __shared__ float xv[CC];
    __shared__ float wt3[3];
    int l = blockIdx.x, b = blockIdx.y, c = threadIdx.x;
    xv[c] = x[((size_t)b * CC + c) * LLQ + l];
    if (c < 3) wt3[c] = wts[b * 3 + c];
    __syncthreads();
    int br = c / 64, oc = c % 64;
    const float* wrow = (br == 0 ? px_w : (br == 1 ? py_w : pz_w)) + (size_t)oc * CC;
    const float* bp   = (br == 0 ? px_b : (br == 1 ? py_b : pz_b));
    float s = 0.f;
    for (int ic = 0; ic < CC; ++ic) s += wrow[ic] * xv[ic];
    xseq[((size_t)b * LLQ + l) * CC + c] = (s + bp[oc]) * wt3[br];
}

// ---------------------------------------------------------------------------
// K3: conv3d(x, off_w1) + bias + exact GELU -> o1 [B,48,L]
// ---------------------------------------------------------------------------
__global__ void conv1_gelu_kernel(const float* __restrict__ x,
                                  const float* __restrict__ w1,
                                  const float* __restrict__ b1,
                                  float* __restrict__ o1) {
    int idx = blockIdx.x * blockDim.x + threadIdx.x;
    const int TOT = BB * 48 * LLQ;
    if (idx >= TOT) return;
    int b  = idx / (48 * LLQ);
    int r  = idx % (48 * LLQ);
    int oc = r / LLQ;
    int l  = r % LLQ;
    int d  = l / (HHH * WWW);
    int h  = (l / WWW) % HHH;
    int w  = l % WWW;
    float s = b1[oc];
    for (int kd = -1; kd <= 1; ++kd) {
        int zd = d + kd; if (zd < 0 || zd >= DD) continue;
        for (int kh = -1; kh <= 1; ++kh) {
            int zh = h + kh; if (zh < 0 || zh >= HHH) continue;
            for (int kw = -1; kw <= 1; ++kw) {
                int zw = w + kw; if (zw < 0 || zw >= WWW) continue;
                int sp = (zd * HHH + zh) * WWW + zw;
                int t  = ((kd + 1) * 3 + (kh + 1)) * 3 + (kw + 1);
                const float* xc = x + (size_t)b * CC * LLQ + sp;
                const float* wc = w1 + (size_t)oc * CC * 27 + t;
                for (int ic = 0; ic < CC; ++ic)
                    s += xc[(size_t)ic * LLQ] * wc[(size_t)ic * 27];
            }
        }
    }
    o1[((size_t)b * 48 + oc) * LLQ + l] = gelu_exact(s);
}

// ---------------------------------------------------------------------------
// K4: conv3d(o1, off_w2) -> offsets [B,L,3] (channel-last already)
// ---------------------------------------------------------------------------
__global__ void conv2_kernel(const float* __restrict__ o1,
                             const float* __restrict__ w2,
                             const float* __restrict__ b2,
                             float* __restrict__ offs) {
    int idx = blockIdx.x * blockDim.x + threadIdx.x;
    const int TOT = BB * LLQ * 3;
    if (idx >= TOT) return;
    int b = idx / (LLQ * 3);
    int r = idx % (LLQ * 3);
    int l = r / 3;
    int j = r % 3;
    int d = l / (HHH * WWW);
    int h = (l / WWW) % HHH;
    int w = l % WWW;
    float s = b2[j];
    for (int kd = -1; kd <= 1; ++kd) {
        int zd = d + kd; if (zd < 0 || zd >= DD) continue;
        for (int kh = -1; kh <= 1; ++kh) {
            int zh = h + kh; if (zh < 0 || zh >= HHH) continue;
            for (int kw = -1; kw <= 1; ++kw) {
                int zw = w + kw; if (zw < 0 || zw >= WWW) continue;
                int sp = (zd * HHH + zh) * WWW + zw;
                int t  = ((kd + 1) * 3 + (kh + 1)) * 3 + (kw + 1);
                const float* oc = o1 + (size_t)b * 48 * LLQ + sp;
                const float* wc = w2 + (size_t)j * 48 * 27 + t;
                for (int ic = 0; ic < 48; ++ic)
                    s += oc[(size_t)ic * LLQ] * wc[(size_t)ic * 27];
            }
        }
    }
    offs[((size_t)b * LLQ + l) * 3 + j] = s;
}

// ---------------------------------------------------------------------------
// K5: trilinear grid-sample (align_corners=False, zero pad) accumulated
// into x_seq.  grid = (LLQ, BB), block = CC.
// ---------------------------------------------------------------------------
__global__ void gridsample_add_kernel(const float* __restrict__ x,
                                      const float* __restrict__ offs,
                                      float* __restrict__ xseq) {
    int l = blockIdx.x, b = blockIdx.y, c = threadIdx.x;
    const float* g = offs + ((size_t)b * LLQ + l) * 3;
    float gx = g[0], gy = g[1], gz = g[2];
    float ix = ((gx + 1.0f) * WWW - 1.0f) * 0.5f;
    float iy = ((gy + 1.0f) * HHH - 1.0f) * 0.5f;
    float iz = ((gz + 1.0f) * DD  - 1.0f) * 0.5f;
    float x0 = floorf(ix), y0 = floorf(iy), z0 = floorf(iz);
    const float* xc = x + ((size_t)b * CC + c) * LLQ;
    float acc = 0.f;
    #pragma unroll
    for (int dz = 0; dz <= 1; ++dz)
    #pragma unroll
    for (int dy = 0; dy <= 1; ++dy)
    #pragma unroll
    for (int dx = 0; dx <= 1; ++dx) {
        float xcc = x0 + dx, ycc = y0 + dy, zcc = z0 + dz;
        float wgt = (1.0f - fabsf(ix - xcc)) * (1.0f - fabsf(iy - ycc)) *
                    (1.0f - fabsf(iz - zcc));
        bool valid = (xcc >= 0.f) && (xcc < (float)WWW) &&
                     (ycc >= 0.f) && (ycc < (float)HHH) &&
                     (zcc >= 0.f) && (zcc < (float)DD);
        int xi = (int)fminf(fmaxf(xcc, 0.f), (float)(WWW - 1));
        int yi = (int)fminf(fmaxf(ycc, 0.f), (float)(HHH - 1));
        int zi = (int)fminf(fmaxf(zcc, 0.f), (float)(DD - 1));
        int idx = (zi * HHH + yi) * WWW + xi;
        acc += xc[idx] * (valid ? wgt : 0.0f);
    }
    xseq[((size_t)b * LLQ + l) * CC + c] += acc;
}

// ---------------------------------------------------------------------------
// WMMA GEMM with TDM-staged weight tile:
//   Out[M,N] = A[M,K(lda)] * W[N,K]^T (+bias)(+softplus)
// Block = 128 threads (4 waves); block computes 64 rows x 16 cols.
// The 16xK weight tile is DMA'd to LDS once by the Tensor Data Mover
// (OOB rows beyond N are zero-filled by TDM), synchronized with
// s_wait_tensorcnt, then each wave runs V_WMMA_F32_16X16X4_F32 over K.
// grid = (MROWS/64, ceil(N/16)).  act: 0=none 1=softplus
// ---------------------------------------------------------------------------
__global__ void gemm_wmma_f32_kernel(const float* __restrict__ A, int lda,
                                     const float* __restrict__ Wt,
                                     const float* __restrict__ bias,
                                     float* __restrict__ Out, int ldo,
                                     int N, int K, int act) {
    __shared__ float wtile[16 * KMAX];   // 24 KB: 16 weight rows, pitch = K

    int lane = threadIdx.x & 31;
    int wave = threadIdx.x >> 5;
    int row0 = blockIdx.x * 64 + wave * 16;
    int col0 = blockIdx.y * 16;
    int mrow = row0 + (lane & 15);
    int ncol = col0 + (lane & 15);
    int kk   = (lane & 16) ? 2 : 0;

    // ---- TDM: DMA the 16xK weight tile into LDS (wave 0 only issues) ----
    if (threadIdx.x < 32) {
        unsigned long long gaddr =
            (unsigned long long)(uintptr_t)(Wt + (size_t)col0 * K);
        unsigned int lds_off = (unsigned int)(uintptr_t)&wtile[0]; // LDS offset = addr[31:0]
        unsigned int rows_rem = (unsigned int)(N - col0);          // valid rows from tile start

        u32x4 g0;
        g0[0] = 1u;                                   // count=1, user mode
        g0[1] = lds_off;                              // lds_addr
        g0[2] = (unsigned int)(gaddr & 0xffffffffull);        // global_addr[31:0]
        g0[3] = (unsigned int)((gaddr >> 32) & 0x1ffffffull)  // global_addr[56:32]
                | (2u << 30);                         // type = 2 (image)

        i32x8 g1;
        g1[0] = (int)(2u << 16);                      // wg_mask=0, data_size=4B
        g1[1] = (int)(((unsigned)K & 0xffffu) << 16); // tensor_dim0[15:0]
        g1[2] = (int)((((unsigned)K >> 16) & 0xffffu)
                | ((rows_rem & 0xffffu) << 16));      // tensor_dim0[31:16] | tensor_dim1[15:0]
        g1[3] = (int)(((rows_rem >> 16) & 0xffffu)
                | (((unsigned)K & 0xffffu) << 16));   // tensor_dim1[31:16] | tile_dim0=K
        g1[4] = 16;                                   // tile_dim1=16, tile_dim2=0
        g1[5] = (int)(unsigned)K;                     // tensor_dim0_stride[31:0]
        g1[6] = 0;                                    // stride0[47:32] | dim1_stride[15:0]
        g1[7] = 0;                                    // dim1_stride[47:16]

        i32x4 g2 = {0, 0, 0, 0};                      // 2-D tile: groups 2/3 unused
        i32x4 g3 = {0, 0, 0, 0};
        i32x8 g4 = {0, 0, 0, 0, 0, 0, 0, 0};          // trailing group (clang-23 form)

        __builtin_amdgcn_tensor_load_to_lds(g0, g1, g2, g3, g4, 0);
        __builtin_amdgcn_s_wait_tensorcnt(0);
    }
    __syncthreads();

    const float* arow = A + (size_t)mrow * lda;
    const float* wrow = &wtile[(lane & 15) * K];      // LDS, rows zero-filled past N

    v8f acc = {};
    for (int k0 = 0; k0 < K; k0 += 4) {
        v2f af, bf;
        af.x = arow[k0 + kk];
        af.y = arow[k0 + kk + 1];
        bf.x = wrow[k0 + kk];
        bf.y = wrow[k0 + kk + 1];
        __builtin_prefetch(arow + k0 + 64, 0, 3);     // global_prefetch_b8, near
        acc = __builtin_amdgcn_wmma_f32_16x16x4_f32(
            /*neg_a=*/false, af, /*neg_b=*/false, bf,
            /*c_mod=*/(short)0, acc, /*reuse_a=*/false, /*reuse_b=*/false);
    }
    int crow = row0 + ((lane & 16) ? 8 : 0);
    if (ncol < N) {
        float bv = bias ? bias[ncol] : 0.0f;
        #pragma unroll
        for (int r = 0; r < 8; ++r) {
            float v = acc[r] + bv;
            if (act == 1) v = (v > 20.0f) ? v : log1pf(expf(v));   // softplus
            Out[(size_t)(crow + r) * ldo + ncol] = v;
        }
    }
}

// ---------------------------------------------------------------------------
// K7: depthwise causal conv1d (k=4, left pad 3) + SiLU over xi = xz[:, :384]
// ---------------------------------------------------------------------------
__global__ void dwconv_silu_kernel(const float* __restrict__ xz,
                                   const float* __restrict__ conv_w,
                                   const float* __restrict__ conv_b,
                                   float* __restrict__ xi_conv) {
    int idx = blockIdx.x * blockDim.x + threadIdx.x;
    const int TOT = MROWS * DIN;
    if (idx >= TOT) return;
    int m   = idx / DIN;
    int dch = idx % DIN;
    int b = m / LLQ, l = m % LLQ;
    float s = conv_b[dch];
    #pragma unroll
    for (int k = 0; k < 4; ++k) {
        int ls = l - 3 + k;
        if (ls >= 0)
            s += conv_w[dch * 4 + k] * xz[((size_t)b * LLQ + ls) * 768 + dch];
    }
    xi_conv[(size_t)m * DIN + dch] = s * sigmoidf_(s);
}

// ---------------------------------------------------------------------------
// K10: selective scan.  grid = BB, block = DIN (one thread per channel,
// h[64] + A-row in registers, B/C vectors staged in LDS per step).
// Also applies +u*D_skip and *silu(z) gating.
// ---------------------------------------------------------------------------
__global__ void scan_kernel(const float* __restrict__ delta,
                            const float* __restrict__ xi_conv,
                            const float* __restrict__ dbl,
                            const float* __restrict__ xz,
                            const float* __restrict__ A_log,
                            const float* __restrict__ D_skip,
                            float* __restrict__ ygate) {
    __shared__ float sB[NST];
    __shared__ float sC[NST];
    int b = blockIdx.x, dch = threadIdx.x;
    float Ar[NST], h[NST];
    #pragma unroll
    for (int n = 0; n < NST; ++n) {
        Ar[n] = -expf(A_log[(size_t)dch * NST + n]);
        h[n] = 0.0f;
    }
    float dskip = D_skip[dch];
    for (int l = 0; l < LLQ; ++l) {
        size_t m = (size_t)b * LLQ + l;
        if (dch < 2 * NST) {
            float v = dbl[m * NDBL + DTR + dch];
            if (dch < NST) sB[dch] = v;
            else           sC[dch - NST] = v;
        }
        __syncthreads();
        float dt = delta[m * DIN + dch];
        float u  = xi_conv[m * DIN + dch];
        float du = dt * u;
        float y = 0.0f;
        #pragma unroll
        for (int n = 0; n < NST; ++n) {
            h[n] = expf(dt * Ar[n]) * h[n] + du * sB[n];
            y += h[n] * sC[n];
        }
        y += u * dskip;
        float z = xz[m * 768 + DIN + dch];
        y *= z * sigmoidf_(z);
        ygate[m * DIN + dch] = y;
        __syncthreads();
    }
}

// ---------------------------------------------------------------------------
// K12a: channel attention gate.  grid = BB, block = CC.
// ---------------------------------------------------------------------------
__global__ void chatt_kernel(const float* __restrict__ feat,
                             const float* __restrict__ ca_w1, const float* __restrict__ ca_b1,
                             const float* __restrict__ ca_w2, const float* __restrict__ ca_b2,
                             float* __restrict__ agate) {
    __shared__ float ps[CC];
    __shared__ float hid[48];
    int b = blockIdx.x, c = threadIdx.x;
    float s = 0.f;
    for (int l = 0; l < LLQ; ++l)
        s += feat[((size_t)b * LLQ + l) * CC + c];
    ps[c] = s * (1.0f / (float)LLQ);
    __syncthreads();
    if (c < 48) {
        float t = ca_b1[c];
        for (int ic = 0; ic < CC; ++ic) t += ca_w1[c * CC + ic] * ps[ic];
        hid[c] = gelu_exact(t);
    }
    __syncthreads();
    float t2 = ca_b2[c];
    for (int ic = 0; ic < 48; ++ic) t2 += ca_w2[c * 48 + ic] * hid[ic];
    agate[b * CC + c] = sigmoidf_(t2);
}

// ---------------------------------------------------------------------------
// K12b: apply gate + transpose [B,L,C] -> [B,C,D,H,W]
// ---------------------------------------------------------------------------
__global__ void finalize_kernel(const float* __restrict__ feat,
                                const float* __restrict__ agate,
                                float* __restrict__ out) {
    int idx = blockIdx.x * blockDim.x + threadIdx.x;
    const int TOT = BB * CC * LLQ;
    if (idx >= TOT) return;
    int b = idx / (CC * LLQ);
    int r = idx % (CC * LLQ);
    int c = r / LLQ;
    int l = r % LLQ;
    out[idx] = feat[((size_t)b * LLQ + l) * CC + c] * agate[b * CC + c];
}

// ---------------------------------------------------------------------------
extern "C" void kernel_launch(void* const* d_in, const int* in_sizes, int n_in,
                              void* d_out, int out_size, void* d_ws, size_t ws_size,
                              hipStream_t stream) {
    const float* x         = (const float*)d_in[0];
    const float* px_w      = (const float*)d_in[1];
    const float* px_b      = (const float*)d_in[2];
    const float* py_w      = (const float*)d_in[3];
    const float* py_b      = (const float*)d_in[4];
    const float* pz_w      = (const float*)d_in[5];
    const float* pz_b      = (const float*)d_in[6];
    const float* wg_w      = (const float*)d_in[7];
    const float* wg_b      = (const float*)d_in[8];
    const float* off_w1    = (const float*)d_in[9];
    const float* off_b1    = (const float*)d_in[10];
    const float* off_w2    = (const float*)d_in[11];
    const float* off_b2    = (const float*)d_in[12];
    const float* in_proj_w = (const float*)d_in[13];
    const float* conv_w    = (const float*)d_in[14];
    const float* conv_b    = (const float*)d_in[15];
    const float* x_proj_w  = (const float*)d_in[16];
    const float* dt_proj_w = (const float*)d_in[17];
    const float* dt_proj_b = (const float*)d_in[18];
    const float* A_log     = (const float*)d_in[19];
    const float* D_skip    = (const float*)d_in[20];
    const float* out_proj_w= (const float*)d_in[21];
    const float* ca_w1     = (const float*)d_in[22];
    const float* ca_b1     = (const float*)d_in[23];
    const float* ca_w2     = (const float*)d_in[24];
    const float* ca_b2     = (const float*)d_in[25];
    float* out = (float*)d_out;

    float* fws = (float*)d_ws;
    size_t off = 0;
    auto carve = [&](size_t n) {
        float* p = fws + off;
        off += (n + 63) & ~(size_t)63;
        return p;
    };
    float* pooled  = carve((size_t)BB * CC);
    float* wts     = carve((size_t)BB * 3);
    float* agate   = carve((size_t)BB * CC);
    float* xseq    = carve((size_t)MROWS * CC);
    float* o1      = carve((size_t)BB * 48 * LLQ);
    float* offs    = carve((size_t)BB * LLQ * 3);
    float* xz      = carve((size_t)MROWS * 768);
    float* xi_conv = carve((size_t)MROWS * DIN);
    float* dbl     = carve((size_t)MROWS * NDBL);
    float* delta   = carve((size_t)MROWS * DIN);
    float* ygate   = carve((size_t)MROWS * DIN);
    float* feat    = carve((size_t)MROWS * CC);

    // 1. pooled mean + softmax branch gate
    pool_gate_kernel<<<BB, CC, 0, stream>>>(x, wg_w, wg_b, pooled, wts);
    // 2. tri-directional projections -> x_seq [B,L,C]
    triproj_kernel<<<dim3(LLQ, BB), CC, 0, stream>>>(x, px_w, px_b, py_w, py_b,
                                                     pz_w, pz_b, wts, xseq);
    // 3. offset branch conv1 + GELU
    conv1_gelu_kernel<<<(BB * 48 * LLQ + 255) / 256, 256, 0, stream>>>(x, off_w1, off_b1, o1);
    // 4. offset branch conv2 -> normalized coords [B,L,3]
    conv2_kernel<<<(BB * LLQ * 3 + 255) / 256, 256, 0, stream>>>(o1, off_w2, off_b2, offs);
    // 5. trilinear grid-sample positional embedding, added into x_seq
    gridsample_add_kernel<<<dim3(LLQ, BB), CC, 0, stream>>>(x, offs, xseq);
    // 6. in_proj GEMM: [4096,192] x [768,192]^T -> xz [4096,768]   (WMMA + TDM)
    gemm_wmma_f32_kernel<<<dim3(MROWS / 64, 768 / 16), 128, 0, stream>>>(
        xseq, CC, in_proj_w, nullptr, xz, 768, 768, CC, 0);
    // 7. depthwise causal conv + SiLU -> xi_conv [4096,384]
    dwconv_silu_kernel<<<(MROWS * DIN + 255) / 256, 256, 0, stream>>>(
        xz, conv_w, conv_b, xi_conv);
    // 8. x_proj GEMM: [4096,384] x [140,384]^T -> dbl [4096,140]   (WMMA + TDM)
    gemm_wmma_f32_kernel<<<dim3(MROWS / 64, (NDBL + 15) / 16), 128, 0, stream>>>(
        xi_conv, DIN, x_proj_w, nullptr, dbl, NDBL, NDBL, DIN, 0);
    // 9. dt_proj GEMM + bias + softplus: [4096,12] x [384,12]^T -> delta
    gemm_wmma_f32_kernel<<<dim3(MROWS / 64, DIN / 16), 128, 0, stream>>>(
        dbl, NDBL, dt_proj_w, dt_proj_b, delta, DIN, DIN, DTR, 1);
    // 10. selective scan (+D skip, *silu(z)) -> ygate [4096,384]
    scan_kernel<<<BB, DIN, 0, stream>>>(delta, xi_conv, dbl, xz, A_log, D_skip, ygate);
    // 11. out_proj GEMM: [4096,384] x [192,384]^T -> feat [4096,192] (WMMA + TDM)
    gemm_wmma_f32_kernel<<<dim3(MROWS / 64, CC / 16), 128, 0, stream>>>(
        ygate, DIN, out_proj_w, nullptr, feat, CC, CC, DIN, 0);
    // 12. channel attention gate + final transpose/store
    chatt_kernel<<<BB, CC, 0, stream>>>(feat, ca_w1, ca_b1, ca_w2, ca_b2, agate);
    finalize_kernel<<<(BB * CC * LLQ + 255) / 256, 256, 0, stream>>>(feat, agate, out);
}